// GNN_Encoder_83820581748771
// MI455X (gfx1250) — compile-verified
//
#include <hip/hip_runtime.h>

// ---------------------------------------------------------------------------
// GATv2 3-layer GNN encoder for MI455X (gfx1250, wave32).
//
// MI455X design choices:
//  * Node-state arrays (xl, xr, out, emax, denom) stay L2-resident (192MB L2);
//    streaming buffers (edge_index, per-edge logits, alpha output) use
//    non-temporal cache policy so they don't evict the node state.
//  * Dense node transforms use V_WMMA_F32_16X16X4_F32 (f32 matrix cores),
//    one wave per 16-node x 16-feature tile; bias folded into C; input ReLU
//    (from the previous layer) fused into the A-matrix load.
//  * Segment softmax via ordered-uint atomicMax + float atomicAdd (L2 atomics).
// ---------------------------------------------------------------------------

#define NNODES 262144

typedef __attribute__((ext_vector_type(2))) float v2f;
typedef __attribute__((ext_vector_type(8))) float v8f;

// Monotone order-preserving float <-> uint mapping (for atomic max on floats).
__device__ __forceinline__ unsigned orderF(float f) {
  unsigned u = __float_as_uint(f);
  return (u & 0x80000000u) ? ~u : (u | 0x80000000u);
}
__device__ __forceinline__ float unorderF(unsigned u) {
  return __uint_as_float((u & 0x80000000u) ? (u ^ 0x80000000u) : ~u);
}
#define ORDER_NEG_INF 0x007FFFFFu  // orderF(-inf)

// ---------------------------------------------------------------------------
// Node transform: xl = act(x)*Wl + bl, xr = act(x)*Wr + br via f32 WMMA.
// A layout (16x4 f32): lanes 0-15 rows M=0..15; VGPR0 = K0 (half 0) / K2
// (half 1), VGPR1 = K1/K3.  C/D: VGPR r -> M=r (lanes 0-15) / M=r+8 (16-31).
// Loads are branch-free: index clamped, value masked with v_cndmask.
// ---------------------------------------------------------------------------
template <int DIN, int DOUT, bool RELU_IN>
__global__ void transform_wmma(const float* __restrict__ x,
                               const float* __restrict__ Wl, const float* __restrict__ bl,
                               const float* __restrict__ Wr, const float* __restrict__ br,
                               float* __restrict__ xl, float* __restrict__ xr,
                               int nTiles) {
  const int wave = threadIdx.x >> 5;
  const int lane = threadIdx.x & 31;
  const int tile = blockIdx.x * (blockDim.x >> 5) + wave;
  if (tile >= nTiles) return;
  const int colBase = blockIdx.y * 16;
  const int half = lane >> 4;
  const int l16  = lane & 15;
  const int col  = colBase + l16;
  const bool colOK = (col < DOUT);
  const int colC = colOK ? col : (DOUT - 1);   // clamped (always in-bounds)

  const float bLv = colOK ? bl[colC] : 0.0f;
  const float bRv = colOK ? br[colC] : 0.0f;
  v8f cl, cr;
#pragma unroll
  for (int r = 0; r < 8; ++r) { cl[r] = bLv; cr[r] = bRv; }

  const float* xrow = x + (size_t)(tile * 16 + l16) * DIN;

#pragma unroll
  for (int k0 = 0; k0 < DIN; k0 += 4) {
    const int ka = k0 + 2 * half;
    const int kb = ka + 1;
    const bool okA = (ka < DIN), okB = (kb < DIN);
    const int kaC = okA ? ka : 0;
    const int kbC = okB ? kb : 0;
    // Unconditional loads from clamped indices, masked by v_cndmask.
    float va = xrow[kaC];
    float vb = xrow[kbC];
    if (RELU_IN) { va = fmaxf(va, 0.0f); vb = fmaxf(vb, 0.0f); }
    float wla = Wl[kaC * DOUT + colC];
    float wlb = Wl[kbC * DOUT + colC];
    float wra = Wr[kaC * DOUT + colC];
    float wrb = Wr[kbC * DOUT + colC];
    v2f a, bL, bR;
    a.x  = okA ? va : 0.0f;
    a.y  = okB ? vb : 0.0f;
    bL.x = (okA && colOK) ? wla : 0.0f;
    bL.y = (okB && colOK) ? wlb : 0.0f;
    bR.x = (okA && colOK) ? wra : 0.0f;
    bR.y = (okB && colOK) ? wrb : 0.0f;
    // 8 args: (neg_a, A, neg_b, B, c_mod, C, reuse_a, reuse_b)
    cl = __builtin_amdgcn_wmma_f32_16x16x4_f32(false, a, false, bL, (short)0, cl, false, false);
    cr = __builtin_amdgcn_wmma_f32_16x16x4_f32(false, a, false, bR, (short)0, cr, false, false);
  }

  if (colOK) {
#pragma unroll
    for (int r = 0; r < 8; ++r) {
      const int row = tile * 16 + r + 8 * half;
      xl[(size_t)row * DOUT + col] = cl[r];
      xr[(size_t)row * DOUT + col] = cr[r];
    }
  }
}

// ---------------------------------------------------------------------------
// Per-layer node-state init: emax=-inf (ordered), denom=0, out=bias.
// ---------------------------------------------------------------------------
template <int DOUT>
__global__ void init_nodes(unsigned* __restrict__ emax, float* __restrict__ denom,
                           float* __restrict__ out, const float* __restrict__ bias, int n) {
  const int i = blockIdx.x * blockDim.x + threadIdx.x;
  if (i >= n) return;
  emax[i]  = ORDER_NEG_INF;
  denom[i] = 0.0f;
#pragma unroll
  for (int f = 0; f < DOUT; ++f) out[(size_t)i * DOUT + f] = bias[f];
}

// Edge endpoints with synthesized self-loops; edge index stream is touch-once
// so load it non-temporally (keep L2 for node state).
__device__ __forceinline__ void edge_sd(const int* __restrict__ ei, int E, int e,
                                        int& s, int& d) {
  if (e < E) {
    s = __builtin_nontemporal_load(ei + e);
    d = __builtin_nontemporal_load(ei + E + e);
  } else {
    s = e - E; d = e - E;
  }
}
__device__ __forceinline__ int edge_d(const int* __restrict__ ei, int E, int e) {
  return (e < E) ? __builtin_nontemporal_load(ei + E + e) : (e - E);
}

// ---------------------------------------------------------------------------
// Pass 1: logit = leaky_relu(xl[src] + xr[dst]) . att ; atomic-max into emax.
// ---------------------------------------------------------------------------
template <int DOUT>
__global__ void edge_logit(const int* __restrict__ ei, int E, int EF,
                           const float* __restrict__ xl, const float* __restrict__ xr,
                           const float* __restrict__ att,
                           float* __restrict__ elog, unsigned* __restrict__ emax) {
  const int e = blockIdx.x * blockDim.x + threadIdx.x;
  if (e >= EF) return;
  int s, d; edge_sd(ei, E, e, s, d);
  const float4* ps = (const float4*)(xl + (size_t)s * DOUT);
  const float4* pd = (const float4*)(xr + (size_t)d * DOUT);
  float acc = 0.0f;
#pragma unroll
  for (int q = 0; q < DOUT / 4; ++q) {
    const float4 a = ps[q];
    const float4 b = pd[q];
    float v;
    v = a.x + b.x; v = v > 0.0f ? v : 0.2f * v; acc += v * att[4 * q + 0];
    v = a.y + b.y; v = v > 0.0f ? v : 0.2f * v; acc += v * att[4 * q + 1];
    v = a.z + b.z; v = v > 0.0f ? v : 0.2f * v; acc += v * att[4 * q + 2];
    v = a.w + b.w; v = v > 0.0f ? v : 0.2f * v; acc += v * att[4 * q + 3];
  }
  __builtin_nontemporal_store(acc, elog + e);
  atomicMax(emax + d, orderF(acc));
}

// ---------------------------------------------------------------------------
// Pass 2: e_exp = exp(logit - max[dst]); atomicAdd into denom.
// ---------------------------------------------------------------------------
__global__ void edge_exp(const int* __restrict__ ei, int E, int EF,
                         float* __restrict__ elog, const unsigned* __restrict__ emax,
                         float* __restrict__ denom) {
  const int e = blockIdx.x * blockDim.x + threadIdx.x;
  if (e >= EF) return;
  const int d = edge_d(ei, E, e);
  const float m = unorderF(emax[d]);
  const float ex = expf(__builtin_nontemporal_load(elog + e) - m);
  __builtin_nontemporal_store(ex, elog + e);
  atomicAdd(denom + d, ex);
}

// ---------------------------------------------------------------------------
// Pass 3: alpha = e_exp / (denom[dst]+1e-16); scatter alpha * xl[src] into out.
// ---------------------------------------------------------------------------
template <int DOUT, bool WRITE_ALPHA>
__global__ void edge_accum(const int* __restrict__ ei, int E, int EF,
                           const float* __restrict__ elog, const float* __restrict__ denom,
                           const float* __restrict__ xl, float* __restrict__ out,
                           float* __restrict__ alphaOut) {
  const int e = blockIdx.x * blockDim.x + threadIdx.x;
  if (e >= EF) return;
  int s, d; edge_sd(ei, E, e, s, d);
  const float alpha = __builtin_nontemporal_load(elog + e) / (denom[d] + 1e-16f);
  if (WRITE_ALPHA) __builtin_nontemporal_store(alpha, alphaOut + e);
  const float* ps = xl + (size_t)s * DOUT;
  float* po = out + (size_t)d * DOUT;
#pragma unroll
  for (int f = 0; f < DOUT; ++f) atomicAdd(po + f, alpha * ps[f]);
}

// edge_index_full as float values: [src_full(EF) | dst_full(EF)]
__global__ void write_edge_index(const int* __restrict__ ei, int E, int EF,
                                 float* __restrict__ dst) {
  const int i = blockIdx.x * blockDim.x + threadIdx.x;
  if (i >= 2 * EF) return;
  int v;
  if (i < EF) { v = (i < E) ? __builtin_nontemporal_load(ei + i) : (i - E); }
  else        { const int e = i - EF;
                v = (e < E) ? __builtin_nontemporal_load(ei + E + e) : (e - E); }
  __builtin_nontemporal_store((float)v, dst + i);
}

// ---------------------------------------------------------------------------
extern "C" void kernel_launch(void* const* d_in, const int* in_sizes, int n_in,
                              void* d_out, int out_size, void* d_ws, size_t ws_size,
                              hipStream_t stream) {
  (void)n_in; (void)out_size; (void)ws_size;
  const int N  = NNODES;
  const int E  = in_sizes[1] / 2;   // 4,194,304
  const int EF = E + N;             // + self loops

  const float* x  = (const float*)d_in[0];
  const int*   ei = (const int*)d_in[1];
  // params, insertion order per layer: Wl, Wr, att, bl, br, bias
  const float* P[18];
  for (int i = 0; i < 18; ++i) P[i] = (const float*)d_in[2 + i];
#define WL(i)   P[6*(i)+0]
#define WR(i)   P[6*(i)+1]
#define ATT(i)  P[6*(i)+2]
#define BL(i)   P[6*(i)+3]
#define BR(i)   P[6*(i)+4]
#define BIAS(i) P[6*(i)+5]

  // Workspace layout (floats). Node-state arrays are L2-resident.
  float*    ws    = (float*)d_ws;
  float*    xl    = ws;                       // N*32 max
  float*    xr    = xl + (size_t)N * 32;      // N*32 max
  float*    elog  = xr + (size_t)N * 32;      // EF (logit, then e_exp)
  unsigned* emax  = (unsigned*)(elog + EF);   // N
  float*    denom = (float*)(emax + N);       // N
  float*    h1    = denom + N;                // N*8  (pre-ReLU)
  float*    h2    = h1 + (size_t)N * 8;       // N*16 (pre-ReLU)

  float* out      = (float*)d_out;
  float* hFinal   = out;                        // N*32
  float* eiOut    = out + (size_t)N * 32;       // 2*EF
  float* alphaOut = eiOut + (size_t)2 * EF;     // EF

  const int nTiles = N / 16;                  // 16384
  const dim3 tBlk(256);                       // 8 waves/block, 1 tile/wave
  const int  tGrdX = nTiles / 8;              // 2048
  const int  nodeB = (N + 255) / 256;
  const int  edgeB = (EF + 255) / 256;

  // ---------------- Layer 0: 3 -> 8 ----------------
  transform_wmma<3, 8, false><<<dim3(tGrdX, 1), tBlk, 0, stream>>>(
      x, WL(0), BL(0), WR(0), BR(0), xl, xr, nTiles);
  init_nodes<8><<<nodeB, 256, 0, stream>>>(emax, denom, h1, BIAS(0), N);
  edge_logit<8><<<edgeB, 256, 0, stream>>>(ei, E, EF, xl, xr, ATT(0), elog, emax);
  edge_exp<<<edgeB, 256, 0, stream>>>(ei, E, EF, elog, emax, denom);
  edge_accum<8, false><<<edgeB, 256, 0, stream>>>(ei, E, EF, elog, denom, xl, h1, nullptr);

  // ---------------- Layer 1: 8 -> 16 (ReLU fused into input load) ---------
  transform_wmma<8, 16, true><<<dim3(tGrdX, 1), tBlk, 0, stream>>>(
      h1, WL(1), BL(1), WR(1), BR(1), xl, xr, nTiles);
  init_nodes<16><<<nodeB, 256, 0, stream>>>(emax, denom, h2, BIAS(1), N);
  edge_logit<16><<<edgeB, 256, 0, stream>>>(ei, E, EF, xl, xr, ATT(1), elog, emax);
  edge_exp<<<edgeB, 256, 0, stream>>>(ei, E, EF, elog, emax, denom);
  edge_accum<16, false><<<edgeB, 256, 0, stream>>>(ei, E, EF, elog, denom, xl, h2, nullptr);

  // ---------------- Layer 2: 16 -> 32 (ReLU fused; writes h + alpha) ------
  transform_wmma<16, 32, true><<<dim3(tGrdX, 2), tBlk, 0, stream>>>(
      h2, WL(2), BL(2), WR(2), BR(2), xl, xr, nTiles);
  init_nodes<32><<<nodeB, 256, 0, stream>>>(emax, denom, hFinal, BIAS(2), N);
  edge_logit<32><<<edgeB, 256, 0, stream>>>(ei, E, EF, xl, xr, ATT(2), elog, emax);
  edge_exp<<<edgeB, 256, 0, stream>>>(ei, E, EF, elog, emax, denom);
  edge_accum<32, true><<<edgeB, 256, 0, stream>>>(ei, E, EF, elog, denom, xl, hFinal, alphaOut);

  // edge_index_full output
  write_edge_index<<<(2 * EF + 255) / 256, 256, 0, stream>>>(ei, E, EF, eiOut);
}